// GCN2_38929583571059
// MI455X (gfx1250) — compile-verified
//
#include <hip/hip_runtime.h>
#include <hip/hip_bf16.h>
#include <math.h>

#define N_NODES 50000
#define N_EDGES 800000
#define IN_DIM  128
#define F_SIZE  64
#define N_GRAPHS 512
#define OUT_DIM 10

typedef __attribute__((ext_vector_type(2))) float v2f;
typedef __attribute__((ext_vector_type(8))) float v8f;
typedef __attribute__((ext_vector_type(4))) int   v4i;

#if defined(__has_builtin)
#if __has_builtin(__builtin_amdgcn_global_load_async_to_lds_b128)
#define HAVE_ASYNC_LDS 1
#endif
#if __has_builtin(__builtin_amdgcn_s_wait_asynccnt)
#define HAVE_WAIT_ASYNC 1
#endif
#endif

// Builtin signature (from hipcc diagnostic): param0 = int4* addrspace(1),
// param1 = int4* addrspace(3), then two immediate ints (offset, cpol).
typedef __attribute__((address_space(1))) v4i as1_v4i;
typedef __attribute__((address_space(3))) v4i as3_v4i;

// 16B global -> LDS copy. Async (ASYNCcnt) path on gfx1250 toolchains that
// expose the builtin; synchronous reg-staged fallback otherwise.
__device__ __forceinline__ void copy16_to_lds(const float* g, float* l) {
#if defined(HAVE_ASYNC_LDS)
  __builtin_amdgcn_global_load_async_to_lds_b128(
      (as1_v4i*)g, (as3_v4i*)l, /*offset=*/0, /*cpol=*/0);
#else
  *(float4*)l = *(const float4*)g;
#endif
}

__device__ __forceinline__ void wait_lds_copies() {
#if defined(HAVE_ASYNC_LDS)
#if defined(HAVE_WAIT_ASYNC)
  __builtin_amdgcn_s_wait_asynccnt(0);
#else
  asm volatile("s_wait_asynccnt 0" ::: "memory");
#endif
#endif
}

// D = A(16x4 f32) * B(4x16 f32) + C(16x16 f32), full fp32 precision.
__device__ __forceinline__ v8f wmma4(v2f a, v2f b, v8f c) {
  return __builtin_amdgcn_wmma_f32_16x16x4_f32(
      /*neg_a=*/false, a, /*neg_b=*/false, b,
      /*c_mod=*/(short)0, c, /*reuse_a=*/false, /*reuse_b=*/false);
}

// ---------------- elementwise / scatter kernels ----------------

__global__ void k_scale_copy(const float* __restrict__ src,
                             float* __restrict__ dst, float scale, size_t n) {
  size_t i = (size_t)blockIdx.x * blockDim.x + threadIdx.x;
  if (i < n) dst[i] = scale * src[i];
}

__global__ void k_fill(float* __restrict__ p, float v, size_t n) {
  size_t i = (size_t)blockIdx.x * blockDim.x + threadIdx.x;
  if (i < n) p[i] = v;
}

// Yout[ind0[e]] += scale * val[e] * Xin[ind1[e]]  (128 feats, 2 per thread)
__global__ void k_spmm_scatter(const int* __restrict__ ind,
                               const float* __restrict__ val,
                               const float* __restrict__ Xin,
                               float* __restrict__ Yout, float scale) {
  size_t idx = (size_t)blockIdx.x * blockDim.x + threadIdx.x;
  if (idx >= (size_t)N_EDGES * (IN_DIM / 2)) return;
  int e = (int)(idx >> 6);
  int f = (int)(idx & 63) * 2;
  int dst = ind[e];                 // indices[0]
  int src = ind[N_EDGES + e];       // indices[1]
  float s = scale * val[e];
  float2 x = *(const float2*)(Xin + (size_t)src * IN_DIM + f);
  float* yp = Yout + (size_t)dst * IN_DIM + f;
  atomicAdd(yp + 0, s * x.x);
  atomicAdd(yp + 1, s * x.y);
}

__global__ void k_deg_edges(const int* __restrict__ ind, float* __restrict__ deg) {
  int e = blockIdx.x * blockDim.x + threadIdx.x;
  if (e < N_EDGES) atomicAdd(deg + ind[N_EDGES + e], 1.0f);  // col = indices[1]
}

__global__ void k_rsqrt_inplace(float* __restrict__ p, size_t n) {
  size_t i = (size_t)blockIdx.x * blockDim.x + threadIdx.x;
  if (i < n) p[i] = rsqrtf(p[i]);   // deg >= 1 always (self loop)
}

// Acc[i] = dis[i]^2 * H[i]   (self-loop contribution)
__global__ void k_self_init(const float* __restrict__ dis,
                            const float* __restrict__ H,
                            float* __restrict__ Acc) {
  size_t idx = (size_t)blockIdx.x * blockDim.x + threadIdx.x;
  if (idx >= (size_t)N_NODES * F_SIZE) return;
  int i = (int)(idx >> 6);
  float d = dis[i];
  Acc[idx] = d * d * H[idx];
}

// Acc[col[e]] += dis[row]*dis[col] * H[row[e]]  (64 feats, 2 per thread)
__global__ void k_gcn_scatter(const int* __restrict__ ind,
                              const float* __restrict__ dis,
                              const float* __restrict__ H,
                              float* __restrict__ Acc) {
  size_t idx = (size_t)blockIdx.x * blockDim.x + threadIdx.x;
  if (idx >= (size_t)N_EDGES * (F_SIZE / 2)) return;
  int e = (int)(idx >> 5);
  int f = (int)(idx & 31) * 2;
  int r = ind[e];                 // row = indices[0] (message source)
  int c = ind[N_EDGES + e];       // col = indices[1] (destination)
  float nrm = dis[r] * dis[c];
  float2 hv = *(const float2*)(H + (size_t)r * F_SIZE + f);
  float* ap = Acc + (size_t)c * F_SIZE + f;
  atomicAdd(ap + 0, nrm * hv.x);
  atomicAdd(ap + 1, nrm * hv.y);
}

__global__ void k_relu_bias(const float* __restrict__ Acc,
                            const float* __restrict__ bias,
                            float* __restrict__ Xout) {
  size_t idx = (size_t)blockIdx.x * blockDim.x + threadIdx.x;
  if (idx >= (size_t)N_NODES * F_SIZE) return;
  int f = (int)(idx & (F_SIZE - 1));
  float v = Acc[idx] + bias[f];
  Xout[idx] = v > 0.0f ? v : 0.0f;
}

// ---------------- WMMA dense GEMMs (LDS-staged A) ----------------
// Block = 128 threads = 4 waves; each wave owns one 16x16 D tile of the
// 16row x 64col block output. A chunks (16x32 f32) are staged in LDS via
// async global->LDS B128 copies (coalesced, VGPR-free), then each wave
// feeds WMMA fragments from LDS.
// LDS row stride 36 floats: rows stay 16B-aligned for B128 writes and
// 36*ln mod 64 spreads the 16 fragment readers across 16 distinct banks.
//
// Fragment layouts (ISA 7.12.2):
//  A 16x4 f32: lane<16 -> {K=k0,k0+1}, lane>=16 -> {K=k0+2,k0+3}, M=lane&15
//  B 4x16 f32: same K split, N=lane&15
//  C/D 16x16:  vgpr r -> row r + 8*(lane>>4), col lane&15

#define KCHUNK 32
#define LDS_STRIDE 36

// H[N,64] = [X | T1 | T2] (N x 384 virtual concat) @ W (384 x 64)
__global__ void k_gemm_cheb(const float* __restrict__ T0,
                            const float* __restrict__ T1,
                            const float* __restrict__ T2,
                            const float* __restrict__ W,
                            float* __restrict__ H) {
  __shared__ alignas(16) float Atile[16 * LDS_STRIDE];
  const int tid = threadIdx.x;
  const int lane = tid & 31;
  const int wave = tid >> 5;
  const int mb = blockIdx.x * 16;     // N_NODES % 16 == 0 -> EXEC all 1s
  const int nb = wave * 16;
  const int ln = lane & 15;
  const int kh = lane >> 4;           // 0 or 1
  const int lrow = tid >> 3;          // loader: 8 threads x 16B per row
  const int lqf = (tid & 7) * 4;
  v8f c = {};
  const float* srcs[3] = {T0, T1, T2};
#pragma unroll 1
  for (int s = 0; s < 3; ++s) {
    const float* A  = srcs[s] + (size_t)mb * IN_DIM;
    const float* Wp = W + (size_t)s * IN_DIM * F_SIZE + nb + ln;
#pragma unroll 1
    for (int kb = 0; kb < IN_DIM; kb += KCHUNK) {
      __syncthreads();   // previous chunk's readers done before overwrite
      copy16_to_lds(A + (size_t)lrow * IN_DIM + kb + lqf,
                    &Atile[lrow * LDS_STRIDE + lqf]);
      wait_lds_copies();
      __syncthreads();
#pragma unroll
      for (int k = 0; k < KCHUNK; k += 4) {
        v2f a, b;
        a[0] = Atile[ln * LDS_STRIDE + k + 2 * kh];
        a[1] = Atile[ln * LDS_STRIDE + k + 2 * kh + 1];
        b[0] = Wp[(size_t)(kb + k + 2 * kh) * F_SIZE];
        b[1] = Wp[(size_t)(kb + k + 2 * kh + 1) * F_SIZE];
        c = wmma4(a, b, c);
      }
    }
  }
  float* Hp = H + (size_t)(mb + 8 * kh) * F_SIZE + nb + ln;
#pragma unroll
  for (int r = 0; r < 8; ++r) Hp[(size_t)r * F_SIZE] = c[r];
}

// H[N,64] = A (N x 64) @ W (64 x 64)
__global__ void k_gemm64(const float* __restrict__ A_,
                         const float* __restrict__ W,
                         float* __restrict__ H) {
  __shared__ alignas(16) float Atile[16 * LDS_STRIDE];
  const int tid = threadIdx.x;
  const int lane = tid & 31;
  const int wave = tid >> 5;
  const int mb = blockIdx.x * 16;
  const int nb = wave * 16;
  const int ln = lane & 15;
  const int kh = lane >> 4;
  const int lrow = tid >> 3;
  const int lqf = (tid & 7) * 4;
  v8f c = {};
  const float* A  = A_ + (size_t)mb * F_SIZE;
  const float* Wp = W + nb + ln;
#pragma unroll 1
  for (int kb = 0; kb < F_SIZE; kb += KCHUNK) {
    __syncthreads();
    copy16_to_lds(A + (size_t)lrow * F_SIZE + kb + lqf,
                  &Atile[lrow * LDS_STRIDE + lqf]);
    wait_lds_copies();
    __syncthreads();
#pragma unroll
    for (int k = 0; k < KCHUNK; k += 4) {
      v2f a, b;
      a[0] = Atile[ln * LDS_STRIDE + k + 2 * kh];
      a[1] = Atile[ln * LDS_STRIDE + k + 2 * kh + 1];
      b[0] = Wp[(size_t)(kb + k + 2 * kh) * F_SIZE];
      b[1] = Wp[(size_t)(kb + k + 2 * kh + 1) * F_SIZE];
      c = wmma4(a, b, c);
    }
  }
  float* Hp = H + (size_t)(mb + 8 * kh) * F_SIZE + nb + ln;
#pragma unroll
  for (int r = 0; r < 8; ++r) Hp[(size_t)r * F_SIZE] = c[r];
}

// ---------------- pooling + head ----------------

__global__ void k_pool_accum(const float* __restrict__ x1,
                             const float* __restrict__ x2,
                             const float* __restrict__ x3,
                             const int* __restrict__ batch,
                             float* __restrict__ pooled) {
  size_t idx = (size_t)blockIdx.x * blockDim.x + threadIdx.x;
  if (idx >= (size_t)N_NODES * F_SIZE) return;
  int i = (int)(idx >> 6);
  int f = (int)(idx & (F_SIZE - 1));
  float v = (x1[idx] + x2[idx] + x3[idx]) * (1.0f / 3.0f);
  atomicAdd(pooled + (size_t)batch[i] * F_SIZE + f, v);
}

__global__ void k_pool_count(const int* __restrict__ batch,
                             float* __restrict__ counts) {
  int i = blockIdx.x * blockDim.x + threadIdx.x;
  if (i < N_NODES) atomicAdd(counts + batch[i], 1.0f);
}

__global__ void k_head(const float* __restrict__ pooled,
                       const float* __restrict__ counts,
                       const float* __restrict__ Wout,
                       const float* __restrict__ bout,
                       float* __restrict__ out) {
  int g = blockIdx.x * blockDim.x + threadIdx.x;
  if (g >= N_GRAPHS) return;
  float inv = 1.0f / fmaxf(counts[g], 1.0f);
  float logit[OUT_DIM];
#pragma unroll
  for (int t = 0; t < OUT_DIM; ++t) logit[t] = bout[t];
  for (int k = 0; k < F_SIZE; ++k) {
    float p = pooled[(size_t)g * F_SIZE + k] * inv;
#pragma unroll
    for (int t = 0; t < OUT_DIM; ++t)
      logit[t] = fmaf(p, Wout[k * OUT_DIM + t], logit[t]);
  }
  float mx = logit[0];
#pragma unroll
  for (int t = 1; t < OUT_DIM; ++t) mx = fmaxf(mx, logit[t]);
  float s = 0.0f;
#pragma unroll
  for (int t = 0; t < OUT_DIM; ++t) { logit[t] = expf(logit[t] - mx); s += logit[t]; }
  float is = 1.0f / s;
#pragma unroll
  for (int t = 0; t < OUT_DIM; ++t) out[(size_t)g * OUT_DIM + t] = logit[t] * is;
}

// ---------------- host orchestration ----------------

static inline unsigned cdiv(size_t a, size_t b) { return (unsigned)((a + b - 1) / b); }

extern "C" void kernel_launch(void* const* d_in, const int* in_sizes, int n_in,
                              void* d_out, int out_size, void* d_ws, size_t ws_size,
                              hipStream_t stream) {
  (void)in_sizes; (void)n_in; (void)out_size; (void)ws_size;
  const float* X    = (const float*)d_in[0];
  const int*   Lind = (const int*)d_in[1];     // [2, N_EDGES] flat
  const float* Lval = (const float*)d_in[2];
  const int*   batch= (const int*)d_in[3];
  const float* W1   = (const float*)d_in[4];
  const float* b1   = (const float*)d_in[5];
  const float* W2   = (const float*)d_in[6];
  const float* b2   = (const float*)d_in[7];
  const float* W3   = (const float*)d_in[8];
  const float* b3   = (const float*)d_in[9];
  const float* Wout = (const float*)d_in[10];
  const float* bout = (const float*)d_in[11];
  float* out = (float*)d_out;

  // workspace layout (floats); aliasing is safe: all reuse happens strictly
  // after the last reader on the single in-order stream.
  float* ws = (float*)d_ws;
  size_t o = 0;
  float* T1     = ws + o; o += (size_t)N_NODES * IN_DIM;   // 6.4M
  float* T2     = ws + o; o += (size_t)N_NODES * IN_DIM;   // 6.4M
  float* h      = ws + o; o += (size_t)N_NODES * F_SIZE;   // 3.2M
  float* x1     = ws + o; o += (size_t)N_NODES * F_SIZE;   // 3.2M
  float* dis    = ws + o; o += 50176;                      // deg -> rsqrt(deg)
  float* pooled = ws + o; o += (size_t)N_GRAPHS * F_SIZE;  // 32768
  float* counts = ws + o; o += N_GRAPHS;                   // 512
  float* x2  = T1;                                 // reuse after gemm_cheb
  float* x3  = T1 + (size_t)N_NODES * F_SIZE;
  float* acc = T2;                                 // reuse after gemm_cheb

  const size_t NXI = (size_t)N_NODES * IN_DIM;
  const size_t NXF = (size_t)N_NODES * F_SIZE;

  // ---- Chebyshev features: T1 = L@X ; T2 = 2*(L@T1) - X ----
  (void)hipMemsetAsync(T1, 0, NXI * sizeof(float), stream);
  k_scale_copy<<<cdiv(NXI, 256), 256, 0, stream>>>(X, T2, -1.0f, NXI);
  k_spmm_scatter<<<cdiv((size_t)N_EDGES * (IN_DIM / 2), 256), 256, 0, stream>>>(
      Lind, Lval, X, T1, 1.0f);
  k_spmm_scatter<<<cdiv((size_t)N_EDGES * (IN_DIM / 2), 256), 256, 0, stream>>>(
      Lind, Lval, T1, T2, 2.0f);

  // ---- GCN symmetric norm: dis = rsqrt(deg), deg includes self loop ----
  k_fill<<<cdiv(N_NODES, 256), 256, 0, stream>>>(dis, 1.0f, N_NODES);
  k_deg_edges<<<cdiv(N_EDGES, 256), 256, 0, stream>>>(Lind, dis);
  k_rsqrt_inplace<<<cdiv(N_NODES, 256), 256, 0, stream>>>(dis, N_NODES);

  // ---- layer 1: h = [X|T1|T2] @ W1 (WMMA), aggregate, relu+bias ----
  k_gemm_cheb<<<N_NODES / 16, 128, 0, stream>>>(X, T1, T2, W1, h);
  k_self_init<<<cdiv(NXF, 256), 256, 0, stream>>>(dis, h, acc);
  k_gcn_scatter<<<cdiv((size_t)N_EDGES * (F_SIZE / 2), 256), 256, 0, stream>>>(
      Lind, dis, h, acc);
  k_relu_bias<<<cdiv(NXF, 256), 256, 0, stream>>>(acc, b1, x1);

  // ---- layer 2 ----
  k_gemm64<<<N_NODES / 16, 128, 0, stream>>>(x1, W2, h);
  k_self_init<<<cdiv(NXF, 256), 256, 0, stream>>>(dis, h, acc);
  k_gcn_scatter<<<cdiv((size_t)N_EDGES * (F_SIZE / 2), 256), 256, 0, stream>>>(
      Lind, dis, h, acc);
  k_relu_bias<<<cdiv(NXF, 256), 256, 0, stream>>>(acc, b2, x2);

  // ---- layer 3 ----
  k_gemm64<<<N_NODES / 16, 128, 0, stream>>>(x2, W3, h);
  k_self_init<<<cdiv(NXF, 256), 256, 0, stream>>>(dis, h, acc);
  k_gcn_scatter<<<cdiv((size_t)N_EDGES * (F_SIZE / 2), 256), 256, 0, stream>>>(
      Lind, dis, h, acc);
  k_relu_bias<<<cdiv(NXF, 256), 256, 0, stream>>>(acc, b3, x3);

  // ---- mean over layers, global mean pool, head + softmax ----
  (void)hipMemsetAsync(pooled, 0,
                       ((size_t)N_GRAPHS * F_SIZE + N_GRAPHS) * sizeof(float),
                       stream);
  k_pool_accum<<<cdiv(NXF, 256), 256, 0, stream>>>(x1, x2, x3, batch, pooled);
  k_pool_count<<<cdiv(N_NODES, 256), 256, 0, stream>>>(batch, counts);
  k_head<<<cdiv(N_GRAPHS, 64), 64, 0, stream>>>(pooled, counts, Wout, bout, out);
}